// EmilySAGE_bond_87703232184757
// MI455X (gfx1250) — compile-verified
//
#include <hip/hip_runtime.h>
#include <hip/hip_bf16.h>

// ---------------------------------------------------------------------------
// SAGEConv (mean aggregation):
//   out = (segment_mean(feature[src] -> dst)) @ W_l^T + b_l + feature @ W_r^T
// N=50000, E=800000, D=128.  50000 % 16 == 0, (E*32) % 256 == 0 -> no tails.
// ---------------------------------------------------------------------------

#define N_NODES 50000
#define N_EDGES 800000
#define D       128

typedef __attribute__((ext_vector_type(2))) float v2f;
typedef __attribute__((ext_vector_type(8))) float v8f;

// ---------------- zero workspace ----------------
__global__ void EmilySAGE_zero_kernel(float* __restrict__ p, int n) {
    int i = blockIdx.x * blockDim.x + threadIdx.x;
    if (i < n) p[i] = 0.0f;
}

// ---------------- edge scatter: agg_sum[dst] += feature[src]; deg[dst] += 1 ----
// one wave32 per edge; lane l handles floats [4l, 4l+4)
__global__ void EmilySAGE_scatter_kernel(const float* __restrict__ feat,
                                         const long long* __restrict__ ei,
                                         float* __restrict__ agg_sum,
                                         float* __restrict__ deg) {
    int idx  = blockIdx.x * blockDim.x + threadIdx.x;
    int edge = idx >> 5;
    int lane = idx & 31;
    if (edge >= N_EDGES) return;
    long long src = ei[edge];
    long long dst = ei[N_EDGES + edge];
    const float4* srow = reinterpret_cast<const float4*>(feat + src * D);
    float4 v = srow[lane];                       // global_load_b128
    float* drow = agg_sum + dst * D + lane * 4;
    atomicAdd(drow + 0, v.x);                    // global_atomic_add_f32 (no-rtn)
    atomicAdd(drow + 1, v.y);
    atomicAdd(drow + 2, v.z);
    atomicAdd(drow + 3, v.w);
    if (lane == 0) atomicAdd(deg + dst, 1.0f);
}

// ---------------- fused normalize + dual GEMM + bias (fp32 WMMA) ------------
// block = 256 threads = 8 waves; block owns a 16-node row tile,
// wave w owns output columns [16w, 16w+16).
#define LDS_STRIDE 132   // 128 + 4 pad: column reads hit distinct banks

__global__ __launch_bounds__(256)
void EmilySAGE_gemm_kernel(const float* __restrict__ feat,
                           const float* __restrict__ Wl,
                           const float* __restrict__ bl,
                           const float* __restrict__ Wr,
                           const float* __restrict__ agg_sum,
                           const float* __restrict__ deg,
                           float* __restrict__ out) {
    __shared__ float aggs[16 * LDS_STRIDE];
    __shared__ float fets[16 * LDS_STRIDE];

    const int row0 = blockIdx.x * 16;
    const int tid  = threadIdx.x;

    // Stage 16x128 tiles of (normalized agg) and feature into LDS.
    // 2048 elements each / 256 threads = 8 per thread.
#pragma unroll
    for (int i = 0; i < 8; ++i) {
        int e = tid + i * 256;
        int m = e >> 7;          // row in tile
        int k = e & 127;         // feature index
        int r = row0 + m;
        float d   = deg[r];
        float inv = 1.0f / fmaxf(d, 1.0f);
        aggs[m * LDS_STRIDE + k] = agg_sum[r * D + k] * inv;
        fets[m * LDS_STRIDE + k] = feat[r * D + k];
    }
    __syncthreads();

    const int wave = tid >> 5;           // 0..7 -> output column tile
    const int lane = tid & 31;
    const int half = lane >> 4;          // 0 or 1
    const int l16  = lane & 15;
    const int n    = wave * 16 + l16;    // output column owned by this lane

    // A (16x4 f32): VGPR0 = K=k0+2*half, VGPR1 = K=k0+2*half+1, M = lane&15
    // B (4x16 f32): mirrored; B[k][n] = W[n*D + k]  (computes X @ W^T)
    v8f acc = {0.f, 0.f, 0.f, 0.f, 0.f, 0.f, 0.f, 0.f};

#pragma unroll
    for (int k0 = 0; k0 < D; k0 += 4) {
        const int ka = k0 + 2 * half;

        v2f a0 = { aggs[l16 * LDS_STRIDE + ka], aggs[l16 * LDS_STRIDE + ka + 1] };
        v2f b0 = { Wl[n * D + ka],              Wl[n * D + ka + 1] };
        acc = __builtin_amdgcn_wmma_f32_16x16x4_f32(
                  false, a0, false, b0, (short)0, acc, false, false);

        v2f a1 = { fets[l16 * LDS_STRIDE + ka], fets[l16 * LDS_STRIDE + ka + 1] };
        v2f b1 = { Wr[n * D + ka],              Wr[n * D + ka + 1] };
        acc = __builtin_amdgcn_wmma_f32_16x16x4_f32(
                  false, a1, false, b1, (short)0, acc, false, false);
    }

    const float bias = bl[n];
    // C/D layout: element v -> row (v + 8*half), column n
#pragma unroll
    for (int v = 0; v < 8; ++v) {
        int r = row0 + v + 8 * half;
        out[r * D + n] = acc[v] + bias;
    }
}

// ---------------------------------------------------------------------------
extern "C" void kernel_launch(void* const* d_in, const int* in_sizes, int n_in,
                              void* d_out, int out_size, void* d_ws, size_t ws_size,
                              hipStream_t stream) {
    const float*     feat = (const float*)d_in[0];
    const long long* ei   = (const long long*)d_in[1];  // int64 [2, E]
    const float*     Wl   = (const float*)d_in[2];
    const float*     bl   = (const float*)d_in[3];
    const float*     Wr   = (const float*)d_in[4];
    float*           out  = (float*)d_out;

    float* agg_sum = (float*)d_ws;                  // N*D floats
    float* deg     = agg_sum + (size_t)N_NODES * D; // N floats

    // 1) zero accumulators (must happen every call: deterministic)
    {
        int n = N_NODES * D + N_NODES;
        int blocks = (n + 255) / 256;
        EmilySAGE_zero_kernel<<<blocks, 256, 0, stream>>>(agg_sum, n);
    }
    // 2) scatter-add edges (one wave32 per edge)
    {
        long long threads = (long long)N_EDGES * 32;
        int blocks = (int)((threads + 255) / 256);  // = 100000 exactly
        EmilySAGE_scatter_kernel<<<blocks, 256, 0, stream>>>(feat, ei, agg_sum, deg);
    }
    // 3) fused normalize + dual fp32-WMMA GEMM + bias
    {
        int blocks = N_NODES / 16;                  // = 3125 exactly
        EmilySAGE_gemm_kernel<<<blocks, 256, 0, stream>>>(feat, Wl, bl, Wr,
                                                          agg_sum, deg, out);
    }
}